// Expression_78280073937529
// MI455X (gfx1250) — compile-verified
//
#include <hip/hip_runtime.h>
#include <math.h>
#include <stdint.h>

typedef int   v4i __attribute__((ext_vector_type(4)));
typedef float v4f __attribute__((ext_vector_type(4)));

// ---------------------------------------------------------------------------
// Kernel 1: zero the per-variable accumulator (d_ws). Must run every call.
// ---------------------------------------------------------------------------
__global__ __launch_bounds__(256) void zero_agg(float* __restrict__ agg, int n)
{
    int i = blockIdx.x * blockDim.x + threadIdx.x;
    if (i < n) agg[i] = 0.0f;
}

// ---------------------------------------------------------------------------
// Kernel 2: per-edge gather + FMA + scatter-add.
//   msg = variable[v,5] * edge_attr[e] + constraint[c,2]
//   agg[v] += msg   (global_atomic_add_f32, no return)
// 4 edges per thread via 128-bit non-temporal loads: the 96 MB edge stream is
// read once (NT keeps it out of L2), while the ~18 MB of gather targets stay
// L2-resident on the 192 MB cache.
// ---------------------------------------------------------------------------
__global__ __launch_bounds__(256) void edge_scatter(
    const int*   __restrict__ c_idx,
    const int*   __restrict__ v_idx,
    const float* __restrict__ edge_attr,
    const float* __restrict__ constraint,   // [n_cons, 5]
    const float* __restrict__ variable,     // [n_vars, 19]
    float*       __restrict__ agg,          // [n_vars]
    int n_edges)
{
    int g  = blockIdx.x * blockDim.x + threadIdx.x;
    int e0 = g * 4;
    if (e0 + 3 < n_edges) {
        v4i c4 = __builtin_nontemporal_load((const v4i*)(c_idx     + e0));
        v4i v4 = __builtin_nontemporal_load((const v4i*)(v_idx     + e0));
        v4f a4 = __builtin_nontemporal_load((const v4f*)(edge_attr + e0));

        // Random gathers — L2 resident.
        float cc0 = constraint[c4.x * 5 + 2];
        float cc1 = constraint[c4.y * 5 + 2];
        float cc2 = constraint[c4.z * 5 + 2];
        float cc3 = constraint[c4.w * 5 + 2];
        float vv0 = variable[v4.x * 19 + 5];
        float vv1 = variable[v4.y * 19 + 5];
        float vv2 = variable[v4.z * 19 + 5];
        float vv3 = variable[v4.w * 19 + 5];

        float m0 = fmaf(vv0, a4.x, cc0);
        float m1 = fmaf(vv1, a4.y, cc1);
        float m2 = fmaf(vv2, a4.z, cc2);
        float m3 = fmaf(vv3, a4.w, cc3);

        atomicAdd(agg + v4.x, m0);
        atomicAdd(agg + v4.y, m1);
        atomicAdd(agg + v4.z, m2);
        atomicAdd(agg + v4.w, m3);
    } else {
        for (int e = e0; e < n_edges; ++e) {
            int   c  = c_idx[e];
            int   v  = v_idx[e];
            float m  = fmaf(variable[v * 19 + 5], edge_attr[e],
                            constraint[c * 5 + 2]);
            atomicAdd(agg + v, m);
        }
    }
}

// ---------------------------------------------------------------------------
// Kernel 3: candidate gather + finalize.
//   out[i] = agg[v] * variable[v,0] + sqrt(|variable[v,3]|),  v = cand[i]
// Stages the cand_mask tile into LDS via the gfx1250 async-to-LDS path
// (global_load_async_to_lds_b32, tracked with ASYNCcnt).
// ---------------------------------------------------------------------------
__global__ __launch_bounds__(256) void finalize_cand(
    const int*   __restrict__ cand,
    const float* __restrict__ variable,
    const float* __restrict__ agg,
    float*       __restrict__ out,
    int n_cand)
{
    __shared__ int lds_cand[256];

    int i      = blockIdx.x * 256 + threadIdx.x;
    int iclamp = (i < n_cand) ? i : (n_cand - 1);   // all lanes issue, store guarded

    // LDS byte address: low 32 bits of the generic address of a __shared__
    // object are the workgroup-relative LDS offset (flat LDS aperture rule).
    unsigned lds_off = (unsigned)(uintptr_t)&lds_cand[threadIdx.x];
    unsigned long long gaddr = (unsigned long long)(uintptr_t)(cand + iclamp);

    asm volatile("global_load_async_to_lds_b32 %0, %1, off"
                 :: "v"(lds_off), "v"(gaddr)
                 : "memory");
    asm volatile("s_wait_asynccnt 0" ::: "memory");

    int v = lds_cand[threadIdx.x];   // each lane reads its own staged slot
    if (i < n_cand) {
        float a  = agg[v];
        float w0 = variable[v * 19 + 0];
        float w3 = variable[v * 19 + 3];
        out[i] = fmaf(a, w0, sqrtf(fabsf(w3)));
    }
}

// ---------------------------------------------------------------------------
// Launch wrapper.
// Inputs (dict order): constraint[f32], variable[f32], cv_edge_index[int, 2xE],
//                      edge_attr[f32], cand_mask[int]
// ---------------------------------------------------------------------------
extern "C" void kernel_launch(void* const* d_in, const int* in_sizes, int n_in,
                              void* d_out, int out_size, void* d_ws, size_t ws_size,
                              hipStream_t stream)
{
    const float* constraint = (const float*)d_in[0];
    const float* variable   = (const float*)d_in[1];
    const int*   cv_edge    = (const int*)d_in[2];
    const float* edge_attr  = (const float*)d_in[3];
    const int*   cand       = (const int*)d_in[4];
    float*       out        = (float*)d_out;

    const int n_edges = in_sizes[3];          // E
    const int n_vars  = in_sizes[1] / 19;     // variable is [n_vars, 19]
    const int n_cand  = out_size;

    const int* c_idx = cv_edge;               // row 0 of [2, E]
    const int* v_idx = cv_edge + n_edges;     // row 1 of [2, E]

    float* agg = (float*)d_ws;                // n_vars floats of scratch

    {
        int blocks = (n_vars + 255) / 256;
        zero_agg<<<blocks, 256, 0, stream>>>(agg, n_vars);
    }
    {
        int groups = (n_edges + 3) / 4;
        int blocks = (groups + 255) / 256;
        edge_scatter<<<blocks, 256, 0, stream>>>(c_idx, v_idx, edge_attr,
                                                 constraint, variable,
                                                 agg, n_edges);
    }
    {
        int blocks = (n_cand + 255) / 256;
        finalize_cand<<<blocks, 256, 0, stream>>>(cand, variable, agg,
                                                  out, n_cand);
    }
}